// OT_Loss_86990267613681
// MI455X (gfx1250) — compile-verified
//
#include <hip/hip_runtime.h>
#include <math.h>

// ---------------------------------------------------------------------------
// Sinkhorn OT loss, N=8192.  cost_map is rank-4:
//   cost[i,j] = r_i + c_j - 2*((x_i x_j + y_i y_j)/Dmax + t_i s_j/Smax)
// so K = exp(-cost/REG) tiles are generated on the fly with
// V_WMMA_F32_16X16X4_F32 + v_exp_f32 (exp2 with log2(e)/REG folded into the
// fragments).  K (256 MB) is never materialized: per matvec pass we read only
// ~0.7 MB of vectors instead of streaming 256 MB from HBM.
// ---------------------------------------------------------------------------

#define N      8192
#define NT     512            // N / 16 tiles per dimension
#define REGf   10.0f
#define M_EPSf 1e-16f
#define L2E    1.4426950408889634f
#define LN2    0.6931471805599453f

typedef __attribute__((ext_vector_type(2))) float v2f;
typedef __attribute__((ext_vector_type(8))) float v8f;

// scalar slots at start of workspace
#define S_DONE 0   // int   : convergence / bad flag
#define S_DMAX 1   // uint  : bits of dist_cost.max()  (non-negative float)
#define S_SMAX 2   // float : score_cost.max()
#define S_SUMS 3   // float : sum(s_scores)
#define SCAL_N 16

// ---------------------------------------------------------------- reductions
__device__ __forceinline__ float blk_reduce(float val, int op, float* red) {
  // blockDim.x must be 1024.  op: 0=max 1=min 2=sum
  int tid = threadIdx.x;
  red[tid] = val;
  __syncthreads();
  for (int sft = 512; sft > 0; sft >>= 1) {
    if (tid < sft) {
      float o = red[tid + sft];
      red[tid] = (op == 0) ? fmaxf(red[tid], o)
               : (op == 1) ? fminf(red[tid], o)
                           : (red[tid] + o);
    }
    __syncthreads();
  }
  float r = red[0];
  __syncthreads();
  return r;
}

// ------------------------------------------------------------------- setup
__global__ __launch_bounds__(1024)
void k_setup(const float* __restrict__ t, const float* __restrict__ s,
             const float* __restrict__ pts, float* __restrict__ scal,
             float* __restrict__ a, float* __restrict__ b,
             float* __restrict__ u, float* __restrict__ v,
             float* __restrict__ nx, float4* __restrict__ Ppk) {
  __shared__ float red[1024];
  int tid = threadIdx.x;
  float maxt = -3.4e38f, mint = 3.4e38f, maxs = -3.4e38f, mins = 3.4e38f, sums = 0.f;
  for (int i = tid; i < N; i += 1024) {
    float tv = t[i], sv = s[i];
    maxt = fmaxf(maxt, tv); mint = fminf(mint, tv);
    maxs = fmaxf(maxs, sv); mins = fminf(mins, sv);
    sums += sv;
  }
  float MT = blk_reduce(maxt, 0, red);
  float mT = blk_reduce(mint, 1, red);
  float MS = blk_reduce(maxs, 0, red);
  float mS = blk_reduce(mins, 1, red);
  float SS = blk_reduce(sums, 2, red);

  float et = 0.f, es = 0.f;
  for (int i = tid; i < N; i += 1024) {
    et += __builtin_amdgcn_exp2f((t[i] - MT) * L2E);
    es += __builtin_amdgcn_exp2f((s[i] - MS) * L2E);
  }
  float SET = blk_reduce(et, 2, red);
  float SES = blk_reduce(es, 2, red);

  for (int i = tid; i < N; i += 1024) {
    a[i] = __builtin_amdgcn_exp2f((t[i] - MT) * L2E) / SET;   // t_prob
    b[i] = __builtin_amdgcn_exp2f((s[i] - MS) * L2E) / SES;   // s_prob
    u[i] = 1.0f / (float)N;
    v[i] = 1.0f / (float)N;
    float x = pts[2 * i], y = pts[2 * i + 1];
    nx[i] = x * x + y * y;
    Ppk[i] = make_float4(x, y, 0.f, 0.f);
  }
  if (tid == 0) {
    ((int*)scal)[S_DONE] = 0;
    ((unsigned*)scal)[S_DMAX] = 0u;
    float md = fmaxf(MT - mS, MS - mT);          // max |t_i - s_j|
    scal[S_SMAX] = md * md;                      // score_cost.max()
    scal[S_SUMS] = SS;
  }
}

// ------------------------------------------------- dist_cost.max() via WMMA
__global__ __launch_bounds__(256)
void k_dmax(const float2* __restrict__ Ppk2, const float* __restrict__ nx,
            float* __restrict__ scal) {
  int lane = threadIdx.x & 31, w = threadIdx.x >> 5;
  int strip = blockIdx.x * (blockDim.x >> 5) + w;      // 16-row strip
  int half = lane >> 4, l16 = lane & 15;
  int irow = strip * 16 + l16;

  v2f af; { float2 t2 = Ppk2[irow * 2 + half]; af.x = t2.x; af.y = t2.y; }
  float nxr[8];
#pragma unroll
  for (int r = 0; r < 8; ++r) nxr[r] = nx[strip * 16 + r + 8 * half];

  float mx = 0.f;
  for (int jt = 0; jt < NT; ++jt) {
    int jc = jt * 16 + l16;
    v2f bf; { float2 t2 = Ppk2[jc * 2 + half]; bf.x = t2.x; bf.y = t2.y; }
    v8f c = {};
    c = __builtin_amdgcn_wmma_f32_16x16x4_f32(false, af, false, bf,
                                              (short)0, c, false, false);
    float nxj = nx[jc];
#pragma unroll
    for (int r = 0; r < 8; ++r)
      mx = fmaxf(mx, nxr[r] + nxj - 2.f * c[r]);
  }
#pragma unroll
  for (int off = 1; off < 32; off <<= 1)
    mx = fmaxf(mx, __shfl_xor(mx, off, 32));
  if (lane == 0)
    atomicMax(((unsigned*)scal) + S_DMAX, __float_as_uint(fmaxf(mx, 0.f)));
}

// ------------------------------------------------------- pack WMMA fragments
__global__ __launch_bounds__(256)
void k_pack(const float* __restrict__ t, const float* __restrict__ s,
            const float* __restrict__ pts, const float* __restrict__ scal,
            float4* __restrict__ Apk, float4* __restrict__ Bpk,
            float* __restrict__ ro, float* __restrict__ co) {
  int i = blockIdx.x * blockDim.x + threadIdx.x;
  if (i >= N) return;
  float Dm = __uint_as_float(((const unsigned*)scal)[S_DMAX]);
  float Sm = scal[S_SMAX];
  float rD = L2E / (REGf * Dm), rS = L2E / (REGf * Sm);
  float x = pts[2 * i], y = pts[2 * i + 1], ti = t[i], si = s[i];
  Apk[i] = make_float4(x * 2.f * rD, y * 2.f * rD, ti * 2.f * rS, 0.f);
  Bpk[i] = make_float4(x, y, si, 0.f);
  ro[i] = -((x * x + y * y) * rD + ti * ti * rS);
  co[i] = -((x * x + y * y) * rD + si * si * rS);
  // exp2(D + ro_i + co_j) == exp(-cost[i,j]/REG)
}

// ------------------------------------------------ KTu[j] = sum_i u_i K[i,j]
__global__ __launch_bounds__(256)
void k_ktu(const float* __restrict__ scal,
           const float2* __restrict__ Apk2, const float2* __restrict__ Bpk2,
           const float* __restrict__ ro, const float* __restrict__ co,
           const float* __restrict__ u, float* __restrict__ KTu) {
  if (((const int*)scal)[S_DONE]) return;
  int lane = threadIdx.x & 31, w = threadIdx.x >> 5;
  int strip = blockIdx.x * (blockDim.x >> 5) + w;      // 16-col strip
  int half = lane >> 4, l16 = lane & 15;
  int jcol = strip * 16 + l16;

  v2f bf; { float2 t2 = Bpk2[jcol * 2 + half]; bf.x = t2.x; bf.y = t2.y; }
  float cj = co[jcol];
  const float4* ro4 = (const float4*)ro;
  const float4* u4  = (const float4*)u;

  float acc = 0.f;
  for (int it = 0; it < NT; ++it) {
    int i0 = it * 16;
    int irow = i0 + l16;
    v2f af; { float2 t2 = Apk2[irow * 2 + half]; af.x = t2.x; af.y = t2.y; }
    v8f c = {};
    c = __builtin_amdgcn_wmma_f32_16x16x4_f32(false, af, false, bf,
                                              (short)0, c, false, false);
    int base = (i0 >> 2) + 2 * half;
    float4 r0 = ro4[base], r1 = ro4[base + 1];
    float4 u0 = u4[base],  u1 = u4[base + 1];
    acc += __builtin_amdgcn_exp2f(c[0] + r0.x + cj) * u0.x;
    acc += __builtin_amdgcn_exp2f(c[1] + r0.y + cj) * u0.y;
    acc += __builtin_amdgcn_exp2f(c[2] + r0.z + cj) * u0.z;
    acc += __builtin_amdgcn_exp2f(c[3] + r0.w + cj) * u0.w;
    acc += __builtin_amdgcn_exp2f(c[4] + r1.x + cj) * u1.x;
    acc += __builtin_amdgcn_exp2f(c[5] + r1.y + cj) * u1.y;
    acc += __builtin_amdgcn_exp2f(c[6] + r1.z + cj) * u1.z;
    acc += __builtin_amdgcn_exp2f(c[7] + r1.w + cj) * u1.w;
  }
  acc += __shfl_xor(acc, 16, 32);   // combine row halves
  if (half == 0) KTu[jcol] = acc;   // unique writer -> deterministic
}

// ------------------------------------------------- Kv[i] = sum_j K[i,j] v_j
__global__ __launch_bounds__(256)
void k_kv(const float* __restrict__ scal,
          const float2* __restrict__ Apk2, const float2* __restrict__ Bpk2,
          const float* __restrict__ ro, const float* __restrict__ co,
          const float* __restrict__ vv, float* __restrict__ Kv) {
  if (((const int*)scal)[S_DONE]) return;
  int lane = threadIdx.x & 31, w = threadIdx.x >> 5;
  int strip = blockIdx.x * (blockDim.x >> 5) + w;      // 16-row strip
  int half = lane >> 4, l16 = lane & 15;
  int irow = strip * 16 + l16;

  v2f af; { float2 t2 = Apk2[irow * 2 + half]; af.x = t2.x; af.y = t2.y; }
  const float4* ro4 = (const float4*)ro;
  float4 r0 = ro4[strip * 4 + 2 * half];
  float4 r1 = ro4[strip * 4 + 2 * half + 1];
  float rowoff[8] = {r0.x, r0.y, r0.z, r0.w, r1.x, r1.y, r1.z, r1.w};

  float acc[8] = {0.f, 0.f, 0.f, 0.f, 0.f, 0.f, 0.f, 0.f};
  for (int jt = 0; jt < NT; ++jt) {
    int jc = jt * 16 + l16;
    v2f bf; { float2 t2 = Bpk2[jc * 2 + half]; bf.x = t2.x; bf.y = t2.y; }
    v8f c = {};
    c = __builtin_amdgcn_wmma_f32_16x16x4_f32(false, af, false, bf,
                                              (short)0, c, false, false);
    float cj = co[jc];
    float vj = vv[jc];
#pragma unroll
    for (int r = 0; r < 8; ++r)
      acc[r] += __builtin_amdgcn_exp2f(c[r] + rowoff[r] + cj) * vj;
  }
#pragma unroll
  for (int r = 0; r < 8; ++r) {
    float sm = acc[r];
#pragma unroll
    for (int off = 1; off < 16; off <<= 1)   // reduce over the 16 cols/half
      sm += __shfl_xor(sm, off, 32);
    acc[r] = sm;
  }
  if (l16 == 0) {
#pragma unroll
    for (int r = 0; r < 8; ++r)
      Kv[strip * 16 + r + 8 * half] = acc[r];  // unique writer
  }
}

// -------------------------------------------------------------- v update
__global__ __launch_bounds__(1024)
void k_updv(const float* __restrict__ scal, const float* __restrict__ b,
            const float* __restrict__ KTu, float* __restrict__ v_tmp) {
  if (((const int*)scal)[S_DONE]) return;
  int i = blockIdx.x * blockDim.x + threadIdx.x;
  if (i < N) v_tmp[i] = b[i] / (KTu[i] + M_EPSf);
}

// ----------------------------------- finite check + commit (or revert/stop)
__global__ __launch_bounds__(1024)
void k_commit(float* __restrict__ scal, const float* __restrict__ a,
              const float* __restrict__ Kv, const float* __restrict__ v_tmp,
              float* __restrict__ u, float* __restrict__ v) {
  __shared__ float red[1024];
  int* done = (int*)scal;
  if (done[S_DONE]) return;
  int tid = threadIdx.x;
  float badf = 0.f;
  for (int i = tid; i < N; i += 1024) {
    float un = a[i] / (Kv[i] + M_EPSf);
    float vt = v_tmp[i];
    if (!(fabsf(un) <= 3.4e38f) || !(fabsf(vt) <= 3.4e38f)) badf = 1.f;
  }
  float bad = blk_reduce(badf, 2, red);
  if (bad > 0.f) {                        // revert (keep old u,v) and stop
    if (tid == 0) done[S_DONE] = 1;
    return;
  }
  for (int i = tid; i < N; i += 1024) {
    u[i] = a[i] / (Kv[i] + M_EPSf);
    v[i] = v_tmp[i];
  }
}

// ------------------------------------------------------- convergence check
__global__ __launch_bounds__(1024)
void k_err(float* __restrict__ scal, const float* __restrict__ b,
           const float* __restrict__ KTu, const float* __restrict__ v) {
  __shared__ float red[1024];
  int* done = (int*)scal;
  if (done[S_DONE]) return;
  float e = 0.f;
  for (int i = threadIdx.x; i < N; i += 1024) {
    float d = b[i] - KTu[i] * v[i];
    e += d * d;
  }
  float E = blk_reduce(e, 2, red);
  if (threadIdx.x == 0 && E <= 1e-9f) done[S_DONE] = 1;
}

// ------------------------------------------------------------- final loss
__global__ __launch_bounds__(1024)
void k_final(const float* __restrict__ scal, const float* __restrict__ s,
             const float* __restrict__ v, float* __restrict__ out) {
  __shared__ float red[1024];
  int tid = threadIdx.x;
  float p = 0.f;
  for (int i = tid; i < N; i += 1024) {
    float beta = REGf * LN2 * __builtin_amdgcn_logf(v[i] + M_EPSf);
    p += s[i] * beta;
  }
  float S1 = blk_reduce(p, 2, red);        // sum(s_d * beta)
  float sc = scal[S_SUMS];
  float denom = sc * sc + 1e-8f;
  float g1 = sc / denom, g2 = S1 / denom;
  float q = 0.f;
  for (int i = tid; i < N; i += 1024) {
    float beta = REGf * LN2 * __builtin_amdgcn_logf(v[i] + M_EPSf);
    q += s[i] * (g1 * beta - g2);          // s * im_grad
  }
  float L = blk_reduce(q, 2, red);
  if (tid == 0) out[0] = L;
}

// ---------------------------------------------------------------- launcher
extern "C" void kernel_launch(void* const* d_in, const int* in_sizes, int n_in,
                              void* d_out, int out_size, void* d_ws, size_t ws_size,
                              hipStream_t stream) {
  const float* t   = (const float*)d_in[0];
  const float* s   = (const float*)d_in[1];
  const float* pts = (const float*)d_in[2];
  float* out = (float*)d_out;

  float* W = (float*)d_ws;
  size_t off = 0;
  float* scal  = W + off; off += SCAL_N;
  float* a     = W + off; off += N;
  float* b     = W + off; off += N;
  float* u     = W + off; off += N;
  float* v     = W + off; off += N;
  float* v_tmp = W + off; off += N;
  float* KTu   = W + off; off += N;
  float* Kv    = W + off; off += N;
  float* ro    = W + off; off += N;
  float* co    = W + off; off += N;
  float* nx    = W + off; off += N;
  float4* Ppk  = (float4*)(W + off); off += 4 * (size_t)N;
  float4* Apk  = (float4*)(W + off); off += 4 * (size_t)N;
  float4* Bpk  = (float4*)(W + off); off += 4 * (size_t)N;

  dim3 passGrid(NT / 8), passBlk(256);   // 8 wave32 per block, 512 strips

  k_setup<<<1, 1024, 0, stream>>>(t, s, pts, scal, a, b, u, v, nx, Ppk);
  k_dmax<<<passGrid, passBlk, 0, stream>>>((const float2*)Ppk, nx, scal);
  k_pack<<<N / 256, 256, 0, stream>>>(t, s, pts, scal, Apk, Bpk, ro, co);

  for (int it = 1; it <= 100; ++it) {
    k_ktu<<<passGrid, passBlk, 0, stream>>>(scal, (const float2*)Apk,
                                            (const float2*)Bpk, ro, co, u, KTu);
    k_updv<<<N / 1024, 1024, 0, stream>>>(scal, b, KTu, v_tmp);
    k_kv<<<passGrid, passBlk, 0, stream>>>(scal, (const float2*)Apk,
                                           (const float2*)Bpk, ro, co, v_tmp, Kv);
    k_commit<<<1, 1024, 0, stream>>>(scal, a, Kv, v_tmp, u, v);
    if (it % 10 == 0) {
      k_ktu<<<passGrid, passBlk, 0, stream>>>(scal, (const float2*)Apk,
                                              (const float2*)Bpk, ro, co, u, KTu);
      k_err<<<1, 1024, 0, stream>>>(scal, b, KTu, v);
    }
  }
  k_final<<<1, 1024, 0, stream>>>(scal, s, v, out);
}